// ComplexLinearAndLeakyReLU_46308337385497
// MI455X (gfx1250) — compile-verified
//
#include <hip/hip_runtime.h>

typedef __attribute__((ext_vector_type(2))) float v2f;
typedef __attribute__((ext_vector_type(8))) float v8f;

#define EPSF 1e-6f
#define NSLOPE 0.2f
#define Bn 8
#define Cn 2048
#define En 256
#define Fn 256
#define TC 16          // c-values per workgroup -> 48 columns = 3 WMMA N-tiles
#define NCOL 48        // columns per workgroup
#define KSTR 260       // padded K stride (floats); 260 mod 64 == 4 -> conflict-free b64

__global__ __launch_bounds__(256)
void fused_complex_linear_leaky(const float* __restrict__ X,
                                const float* __restrict__ J,
                                const float* __restrict__ A,
                                const float* __restrict__ Bw,
                                const float* __restrict__ Cw,
                                const float* __restrict__ W,
                                float* __restrict__ out)
{
    extern __shared__ float lds[];
    float* La = lds;                     // [48][260] a-vectors, column(n)-major
    float* Lb = lds + NCOL * KSTR;       // [48][260] b-vectors
    float* Lc = lds + 2 * NCOL * KSTR;   // [48][260] c-vectors
    float* Yl = lds;                     // aliases La after stage-1 barrier

    const int tid  = threadIdx.x;
    const int lane = tid & 31;
    const int wave = tid >> 5;
    const int bb   = blockIdx.y;         // batch
    const int cbase = blockIdx.x * TC;   // c-tile origin

    // ---------------- Phase 0: per-point basis + rotation -> LDS ----------------
    // 256 threads: thread = e, loop over the 16 c's in the tile.
    for (int cc = 0; cc < TC; ++cc) {
        const int e = tid;
        const long long base = (((long long)(bb * Cn + cbase + cc)) * En + e) * 3;
        const float jx = J[base + 0], jy = J[base + 1], jz = J[base + 2];
        const float x0 = X[base + 0], x1 = X[base + 1], x2 = X[base + 2];

        const float inv  = 1.0f / (sqrtf(jx*jx + jy*jy + jz*jz) + EPSF);
        const float nJx = jx * inv, nJy = jy * inv, nJz = jz * inv;
        const float Uz  = -(nJx*nJx + nJy*nJy) / (nJz + EPSF);
        const float inv2 = 1.0f / (sqrtf(nJx*nJx + nJy*nJy + Uz*Uz) + EPSF);
        const float nUx = nJx * inv2, nUy = nJy * inv2, nUz = Uz * inv2;
        const float nVx = nUy*nJz - nUz*nJy;
        const float nVy = nUz*nJx - nUx*nJz;
        const float nVz = nUx*nJy - nUy*nJx;

        // rtx_i = R^T X  (R rows = nU,nV,nJ)
        const float r0 = nUx*x0 + nVx*x1 + nJx*x2;
        const float r1 = nUy*x0 + nVy*x1 + nJy*x2;
        const float r2 = nUz*x0 + nVz*x1 + nJz*x2;

        // a = col0*r0 + col1*r1 ; b = col1*r0 - col0*r1 ; c = col2*r2
        const float a0 = nUx*r0 + nUy*r1, a1 = nVx*r0 + nVy*r1, a2 = nJx*r0 + nJy*r1;
        const float b0 = nUy*r0 - nUx*r1, b1 = nVy*r0 - nVx*r1, b2 = nJy*r0 - nJx*r1;
        const float cv0 = nUz*r2,         cv1 = nVz*r2,         cv2 = nJz*r2;

        // column n = i*16 + cc, element k = e  ->  L[n*KSTR + e]
        float* pa = La + cc * KSTR + e;
        pa[0] = a0;  pa[16 * KSTR] = a1;  pa[32 * KSTR] = a2;
        float* pb = Lb + cc * KSTR + e;
        pb[0] = b0;  pb[16 * KSTR] = b1;  pb[32 * KSTR] = b2;
        float* pc = Lc + cc * KSTR + e;
        pc[0] = cv0; pc[16 * KSTR] = cv1; pc[32 * KSTR] = cv2;
    }
    __syncthreads();

    // ---------------- Stage 1: Y = A@a + Bw@b + Cw@c  (K = e, 256) -------------
    const int rowA  = lane & 15;            // M row (A-frag) / N col (B-frag)
    const int khalf = (lane >> 4) << 1;     // lanes 16-31 hold K+2,K+3
    const int rhi   = (lane >> 4) << 3;     // C/D: lanes 16-31 hold M+8

    v8f acc[2][3] = {};                     // x = Y tiles, stay in registers
    for (int k = 0; k < En; k += 4) {
        const int koff = k + khalf;
        v2f wa[2], wb[2], wc[2];
#pragma unroll
        for (int m = 0; m < 2; ++m) {
            const int f = (wave * 2 + m) * 16 + rowA;
            wa[m] = *(const v2f*)(A  + f * En + koff);
            wb[m] = *(const v2f*)(Bw + f * En + koff);
            wc[m] = *(const v2f*)(Cw + f * En + koff);
        }
        v2f fa[3], fb[3], fc[3];
#pragma unroll
        for (int t = 0; t < 3; ++t) {
            const int n = t * 16 + rowA;
            fa[t] = *(const v2f*)(La + n * KSTR + koff);   // single ds_load_b64
            fb[t] = *(const v2f*)(Lb + n * KSTR + koff);
            fc[t] = *(const v2f*)(Lc + n * KSTR + koff);
        }
#pragma unroll
        for (int m = 0; m < 2; ++m)
#pragma unroll
            for (int t = 0; t < 3; ++t) {
                acc[m][t] = __builtin_amdgcn_wmma_f32_16x16x4_f32(
                    false, wa[m], false, fa[t], (short)0, acc[m][t], false, false);
                acc[m][t] = __builtin_amdgcn_wmma_f32_16x16x4_f32(
                    false, wb[m], false, fb[t], (short)0, acc[m][t], false, false);
                acc[m][t] = __builtin_amdgcn_wmma_f32_16x16x4_f32(
                    false, wc[m], false, fc[t], (short)0, acc[m][t], false, false);
            }
    }
    __syncthreads();   // everyone done reading La/Lb/Lc

    // Spill Y into LDS: element (row f, col n) at Yl[n*KSTR + f]; reuses La region.
#pragma unroll
    for (int m = 0; m < 2; ++m) {
        const int f0 = (wave * 2 + m) * 16;
#pragma unroll
        for (int t = 0; t < 3; ++t) {
            const int n = t * 16 + rowA;
#pragma unroll
            for (int v = 0; v < 8; ++v)
                Yl[n * KSTR + f0 + v + rhi] = acc[m][t][v];
        }
    }
    __syncthreads();

    // ---------------- Stage 2: d = W @ Y  (K = g, 256) -------------------------
    v8f dacc[2][3] = {};
    for (int g = 0; g < Fn; g += 4) {
        const int koff = g + khalf;
        v2f ww[2];
#pragma unroll
        for (int m = 0; m < 2; ++m) {
            const int f = (wave * 2 + m) * 16 + rowA;
            ww[m] = *(const v2f*)(W + f * Fn + koff);
        }
        v2f fy[3];
#pragma unroll
        for (int t = 0; t < 3; ++t) {
            const int n = t * 16 + rowA;
            fy[t] = *(const v2f*)(Yl + n * KSTR + koff);
        }
#pragma unroll
        for (int m = 0; m < 2; ++m)
#pragma unroll
            for (int t = 0; t < 3; ++t)
                dacc[m][t] = __builtin_amdgcn_wmma_f32_16x16x4_f32(
                    false, ww[m], false, fy[t], (short)0, dacc[m][t], false, false);
    }

    // ---------------- Fusion: dot/dns over i (3 N-tiles) entirely in-lane ------
#pragma unroll
    for (int m = 0; m < 2; ++m) {
#pragma unroll
        for (int v = 0; v < 8; ++v) {
            const int f = (wave * 2 + m) * 16 + v + rhi;
            float dot = 0.0f, dns = 0.0f;
#pragma unroll
            for (int t = 0; t < 3; ++t) {
                dot += acc[m][t][v] * dacc[m][t][v];
                dns += dacc[m][t][v] * dacc[m][t][v];
            }
            // dot>=0: x_out = x ; else x_out = x - (1-NSLOPE)*dot/(dns+eps)*d
            const float scale = (dot >= 0.0f) ? 0.0f
                               : (1.0f - NSLOPE) * dot / (dns + EPSF);
#pragma unroll
            for (int t = 0; t < 3; ++t) {
                const long long o =
                    (((long long)(bb * Fn + f)) * 3 + t) * Cn + cbase + rowA;
                out[o] = acc[m][t][v] - scale * dacc[m][t][v];
            }
        }
    }
}

extern "C" void kernel_launch(void* const* d_in, const int* in_sizes, int n_in,
                              void* d_out, int out_size, void* d_ws, size_t ws_size,
                              hipStream_t stream) {
    const float* X  = (const float*)d_in[0];
    const float* J  = (const float*)d_in[1];
    const float* A  = (const float*)d_in[2];
    const float* Bw = (const float*)d_in[3];
    const float* Cw = (const float*)d_in[4];
    const float* W  = (const float*)d_in[5];
    float* out = (float*)d_out;

    const size_t shmem = (size_t)3 * NCOL * KSTR * sizeof(float); // ~146 KB < 320 KB
    hipFuncSetAttribute(reinterpret_cast<const void*>(&fused_complex_linear_leaky),
                        hipFuncAttributeMaxDynamicSharedMemorySize, (int)shmem);

    dim3 grid(Cn / TC, Bn);   // 128 x 8 = 1024 workgroups
    fused_complex_linear_leaky<<<grid, 256, shmem, stream>>>(X, J, A, Bw, Cw, W, out);
}